// GIN_43671227466212
// MI455X (gfx1250) — compile-verified
//
#include <hip/hip_runtime.h>

typedef __attribute__((ext_vector_type(2))) float v2f;
typedef __attribute__((ext_vector_type(8))) float v8f;

// ---------------------------------------------------------------------------
// GEMM: out[r][n] = relu( sum_k A[r][k] * W[n][k] + bias[n] )
// A: [nrows x 128] row-major, W: [128 x 128] row-major (used as B = W^T),
// out: [nrows x 128]. One wave -> one 16x16 tile via V_WMMA_F32_16X16X4_F32,
// 32 K-steps of K=4. 8 waves per block cover the 8 column tiles of a 16-row
// stripe. nrows must be a multiple of 16 (50000 = 3125*16).
// ---------------------------------------------------------------------------
__global__ __launch_bounds__(256) void gin_gemm_relu_f32(
    const float* __restrict__ A, const float* __restrict__ W,
    const float* __restrict__ bias, float* __restrict__ out, int nrows)
{
  const int lane = threadIdx.x & 31;
  const int wave = threadIdx.x >> 5;     // column tile 0..7
  const int row0 = blockIdx.x * 16;
  if (row0 >= nrows) return;
  const int n0   = wave * 16;
  const int half = lane >> 4;            // 0: K pair {0,1}, 1: K pair {2,3}
  const int l    = lane & 15;

  // A element layout (16x4 f32): lane l (l<16): a = {A[l][k0],   A[l][k0+1]}
  //                              lane l+16    : a = {A[l][k0+2], A[l][k0+3]}
  // B element layout (4x16 f32): lane l (l<16): b = {B[k0][l],   B[k0+1][l]}
  //                              lane l+16    : b = {B[k0+2][l], B[k0+3][l]}
  // with B[k][n] = W[n][k]  => contiguous float2 from row (n0+l) of W.
  const float* aPtr = A + (size_t)(row0 + l) * 128 + 2 * half;
  const float* wPtr = W + (size_t)(n0 + l)   * 128 + 2 * half;

  v8f c = {};
#pragma unroll
  for (int kk = 0; kk < 32; ++kk) {
    const int k0 = kk * 4;
    v2f a = *(const v2f*)(aPtr + k0);
    v2f b = *(const v2f*)(wPtr + k0);
    // (neg_a, A, neg_b, B, c_mod, C, reuse_a, reuse_b)
    c = __builtin_amdgcn_wmma_f32_16x16x4_f32(
        false, a, false, b, (short)0, c, false, false);
  }

  const float bv = bias[n0 + l];
#pragma unroll
  for (int r = 0; r < 8; ++r) {
    const int m = r + half * 8;          // C/D layout: VGPR r holds M=r / M=r+8
    float v = c[r] + bv;
    v = v > 0.0f ? v : 0.0f;
    out[(size_t)(row0 + m) * 128 + n0 + l] = v;
  }
}

// ---------------------------------------------------------------------------
// Edge scatter: n[dst[e]][:] += h[src[e]][:]. One wave per edge; each lane
// handles 4 consecutive floats (float4 gather + 4 native f32 global atomics,
// relaxed/agent scope -> GLOBAL_ATOMIC_ADD_F32 without return, L2-resident).
// ---------------------------------------------------------------------------
__global__ __launch_bounds__(256) void gin_scatter_add(
    const float* __restrict__ h, const long long* __restrict__ src,
    const long long* __restrict__ dst, float* __restrict__ nacc, int nedges)
{
  const int e = blockIdx.x * 8 + (threadIdx.x >> 5);
  if (e >= nedges) return;
  const int lane = threadIdx.x & 31;
  const long long s = src[e];
  const long long d = dst[e];

  const float4 v = ((const float4*)(h + (size_t)s * 128))[lane];
  float* np = nacc + (size_t)d * 128 + lane * 4;
  __hip_atomic_fetch_add(np + 0, v.x, __ATOMIC_RELAXED, __HIP_MEMORY_SCOPE_AGENT);
  __hip_atomic_fetch_add(np + 1, v.y, __ATOMIC_RELAXED, __HIP_MEMORY_SCOPE_AGENT);
  __hip_atomic_fetch_add(np + 2, v.z, __ATOMIC_RELAXED, __HIP_MEMORY_SCOPE_AGENT);
  __hip_atomic_fetch_add(np + 3, v.w, __ATOMIC_RELAXED, __HIP_MEMORY_SCOPE_AGENT);
}

// ---------------------------------------------------------------------------
// h2 = 1 + eps*h + n   (written over n in workspace, so GEMM2 has no RAW
// hazard against its own output buffer)
// ---------------------------------------------------------------------------
__global__ __launch_bounds__(256) void gin_combine(
    const float* __restrict__ h, float* __restrict__ nacc,
    const float* __restrict__ eps, int total)
{
  const int i = blockIdx.x * blockDim.x + threadIdx.x;
  if (i < total) {
    const float e = eps[0];
    nacc[i] = 1.0f + e * h[i] + nacc[i];
  }
}

// ---------------------------------------------------------------------------
// inputs: feats, src, dst, W_f, b_f, W_phy, b_phy, eps
// ---------------------------------------------------------------------------
extern "C" void kernel_launch(void* const* d_in, const int* in_sizes, int n_in,
                              void* d_out, int out_size, void* d_ws, size_t ws_size,
                              hipStream_t stream)
{
  const float*     feats = (const float*)d_in[0];
  const long long* src   = (const long long*)d_in[1];
  const long long* dst   = (const long long*)d_in[2];
  const float*     W_f   = (const float*)d_in[3];
  const float*     b_f   = (const float*)d_in[4];
  const float*     W_phy = (const float*)d_in[5];
  const float*     b_phy = (const float*)d_in[6];
  const float*     eps   = (const float*)d_in[7];

  float* out  = (float*)d_out;
  float* nbuf = (float*)d_ws;            // [N x 128] accumulator, then h2

  const int DIM = 128;
  const int N   = in_sizes[0] / DIM;     // 50000
  const int E   = in_sizes[1];           // 625000
  const int total = N * DIM;

  // n = 0
  hipMemsetAsync(nbuf, 0, (size_t)total * sizeof(float), stream);

  // h = relu(feats @ W_f^T + b_f)   -> d_out
  gin_gemm_relu_f32<<<dim3((N + 15) / 16), dim3(256), 0, stream>>>(
      feats, W_f, b_f, out, N);

  // n[dst] += h[src]
  gin_scatter_add<<<dim3((E + 7) / 8), dim3(256), 0, stream>>>(
      out, src, dst, nbuf, E);

  // h2 = 1 + eps*h + n              -> nbuf
  gin_combine<<<dim3((total + 255) / 256), dim3(256), 0, stream>>>(
      out, nbuf, eps, total);

  // out = relu(h2 @ W_phy^T + b_phy) -> d_out
  gin_gemm_relu_f32<<<dim3((N + 15) / 16), dim3(256), 0, stream>>>(
      nbuf, W_phy, b_phy, out, N);
}